// ChamferLoss_45406394253979
// MI455X (gfx1250) — compile-verified
//
#include <hip/hip_runtime.h>

typedef __attribute__((ext_vector_type(2))) float v2f;
typedef __attribute__((ext_vector_type(8))) float v8f;

#define WAVES_PER_BLOCK 4

// ---------------------------------------------------------------------------
// Pre-pack B fragments: for column tile t and lane l (32 lanes per 16-col
// tile), emit {bv.x, bv.y, |q|^2, 0} where bv is the V_WMMA_F32_16X16X4_F32
// B-operand element pair for that lane (lanes 0-15 hold K=0,1; lanes 16-31
// hold K=2,3 with the K=3 slot zeroed).  One aligned float4 per (tile,lane).
// ---------------------------------------------------------------------------
__global__ __launch_bounds__(256) void pack_cols_kernel(
    const float* __restrict__ Q,   // [B, ncols, 3]
    float4* __restrict__ out,      // [B, 2*ncols]
    int ncols)
{
  const int tid  = blockIdx.x * 256 + threadIdx.x;  // 0 .. 2*ncols-1
  const int b    = blockIdx.y;
  const int lane = tid & 31;
  const int tile = tid >> 5;
  const int col  = tile * 16 + (lane & 15);
  const bool hi  = lane >= 16;

  const float* Qb = Q + (size_t)b * ncols * 3;
  const float qx = Qb[col * 3 + 0];
  const float qy = Qb[col * 3 + 1];
  const float qz = Qb[col * 3 + 2];

  float4 o;
  o.x = hi ? qz : qx;
  o.y = hi ? 0.0f : qy;
  o.z = qx * qx + qy * qy + qz * qz;
  o.w = 0.0f;
  out[(size_t)b * (2 * ncols) + tid] = o;
}

// ---------------------------------------------------------------------------
// Cross-lane min reduce (16 lanes per half) + store 16 row results.
// ---------------------------------------------------------------------------
__device__ __forceinline__ void reduce_and_store(
    float* rmin, float p2own, float* __restrict__ dst, int l15, bool hi)
{
#pragma unroll
  for (int k = 0; k < 8; ++k) {
    float v = rmin[k];
    v = fminf(v, __shfl_xor(v, 1, 32));
    v = fminf(v, __shfl_xor(v, 2, 32));
    v = fminf(v, __shfl_xor(v, 4, 32));
    v = fminf(v, __shfl_xor(v, 8, 32));
    rmin[k] = v;
  }
  // Hoist shuffles out of the divergent region (bpermute from disabled lanes
  // returns 0): every lane gathers the 8 |p|^2 values of its half's rows.
  const int roff = hi ? 8 : 0;
  float p2k[8];
#pragma unroll
  for (int k = 0; k < 8; ++k) p2k[k] = __shfl(p2own, k + roff, 32);

  if (l15 == 0) {
#pragma unroll
    for (int k = 0; k < 8; ++k) {
      const float d = p2k[k] + rmin[k];
      dst[roff + k] = d > 0.0f ? d : 0.0f;
    }
  }
}

// ---------------------------------------------------------------------------
// For each row p of P (per batch b):
//   out[b*nrows + p] = max(0, |p|^2 + min_q(|q|^2 - 2 p.q))
// Each wave owns TWO 16-row tiles (two per-wave-constant A fragments) and
// loops over all pre-packed column tiles; one b128 load feeds two WMMAs.
// ---------------------------------------------------------------------------
__global__ __launch_bounds__(32 * WAVES_PER_BLOCK) void chamfer_rowmin_kernel(
    const float* __restrict__ P,        // [B, nrows, 3]
    const float4* __restrict__ Qpack,   // [B, 2*ncols] packed B-frags + |q|^2
    float* __restrict__ out,            // [B, nrows]
    int nrows, int ncols)
{
  const int b    = blockIdx.y;
  const int lane = threadIdx.x & 31;
  const int wave = threadIdx.x >> 5;
  const int rowBase = (blockIdx.x * WAVES_PER_BLOCK + wave) * 32;

  const float*  __restrict__ Pb = P + (size_t)b * nrows * 3;
  const float4* __restrict__ Qp = Qpack + (size_t)b * (2 * ncols) + lane;

  const int  l15 = lane & 15;
  const bool hi  = lane >= 16;   // lanes 16..31 hold K = 2,3 of A/B fragments

  // Two A fragments (rows rowBase..+15 and rowBase+16..+31).
  const int r0 = rowBase + l15;
  const int r1 = rowBase + 16 + l15;
  const float p0x = Pb[r0 * 3 + 0], p0y = Pb[r0 * 3 + 1], p0z = Pb[r0 * 3 + 2];
  const float p1x = Pb[r1 * 3 + 0], p1y = Pb[r1 * 3 + 1], p1z = Pb[r1 * 3 + 2];
  const float p2own0 = p0x * p0x + p0y * p0y + p0z * p0z;
  const float p2own1 = p1x * p1x + p1y * p1y + p1z * p1z;
  v2f a0; a0.x = hi ? p0z : p0x; a0.y = hi ? 0.0f : p0y;
  v2f a1; a1.x = hi ? p1z : p1x; a1.y = hi ? 0.0f : p1y;

  float rmin0[8], rmin1[8];
#pragma unroll
  for (int k = 0; k < 8; ++k) { rmin0[k] = 3.0e38f; rmin1[k] = 3.0e38f; }

  const int ntiles = ncols >> 4;
#pragma unroll 2
  for (int t = 0; t < ntiles; ++t) {
    const float4 qp = Qp[(size_t)t * 32];
    __builtin_prefetch(&Qp[(size_t)(t + 2) * 32], 0, 3);  // speculative, branchless

    v2f bv; bv.x = qp.x; bv.y = qp.y;
    const v8f z = {0.f, 0.f, 0.f, 0.f, 0.f, 0.f, 0.f, 0.f};
    const v8f c0 = __builtin_amdgcn_wmma_f32_16x16x4_f32(
        false, a0, false, bv, (short)0, z, false, false);
    const v8f c1 = __builtin_amdgcn_wmma_f32_16x16x4_f32(
        false, a1, false, bv, (short)0, z, false, false);

#pragma unroll
    for (int k = 0; k < 8; ++k) {
      rmin0[k] = fminf(rmin0[k], __builtin_fmaf(c0[k], -2.0f, qp.z));
      rmin1[k] = fminf(rmin1[k], __builtin_fmaf(c1[k], -2.0f, qp.z));
    }
  }

  float* dst = out + (size_t)b * nrows + rowBase;
  reduce_and_store(rmin0, p2own0, dst, l15, hi);
  reduce_and_store(rmin1, p2own1, dst + 16, l15, hi);
}

// Deterministic per-batch sum: block bx reduces in[bx*n .. bx*n+n).
__global__ __launch_bounds__(256) void rowsum_kernel(
    const float* __restrict__ in, float* __restrict__ out, int n)
{
  __shared__ float s[256];
  const float* src = in + (size_t)blockIdx.x * n;
  float acc = 0.0f;
  for (int i = threadIdx.x; i < n; i += 256) acc += src[i];
  s[threadIdx.x] = acc;
  __syncthreads();
  for (int off = 128; off > 0; off >>= 1) {
    if ((int)threadIdx.x < off) s[threadIdx.x] += s[threadIdx.x + off];
    __syncthreads();
  }
  if (threadIdx.x == 0) out[blockIdx.x] = s[0];
}

// sums[0..B): per-batch p2g sums ; sums[B..2B): per-batch g2p sums.
__global__ void finalize_kernel(const float* __restrict__ sums,
                                float* __restrict__ out, int B, int N, int M)
{
  if (threadIdx.x == 0 && blockIdx.x == 0) {
    float m1 = 0.0f, m2 = 0.0f;
    for (int b = 0; b < B; ++b) {
      m1 += sums[b] / (float)N;
      m2 += sums[B + b] / (float)M;
    }
    m1 /= (float)B;
    m2 /= (float)B;
    out[0] = m1 + m2;
    out[1] = m1;
    out[2] = m2;
  }
}

extern "C" void kernel_launch(void* const* d_in, const int* in_sizes, int n_in,
                              void* d_out, int out_size, void* d_ws, size_t ws_size,
                              hipStream_t stream) {
  (void)n_in; (void)out_size; (void)ws_size;
  const float* points = (const float*)d_in[0];  // [B, N, 3]
  const float* gts    = (const float*)d_in[1];  // [B, M, 3]

  const int B = 4;                       // fixed by the reference setup
  const int N = in_sizes[0] / (B * 3);   // 8192
  const int M = in_sizes[1] / (B * 3);   // 8192

  float* ws = (float*)d_ws;
  float4* gpack = (float4*)ws;                       // B * 2M float4
  float4* ppack = gpack + (size_t)B * 2 * M;         // B * 2N float4
  float*  p2g   = (float*)(ppack + (size_t)B * 2 * N);  // B*N
  float*  g2p   = p2g + (size_t)B * N;               // B*M
  float*  sums  = g2p + (size_t)B * M;               // 2*B

  // Pack B-operand fragments for both directions.
  pack_cols_kernel<<<dim3((2 * M) / 256, B), 256, 0, stream>>>(gts, gpack, M);
  pack_cols_kernel<<<dim3((2 * N) / 256, B), 256, 0, stream>>>(points, ppack, N);

  dim3 blk(32 * WAVES_PER_BLOCK);

  // Pass 1: each point -> nearest gt.
  chamfer_rowmin_kernel<<<dim3(N / (32 * WAVES_PER_BLOCK), B), blk, 0, stream>>>(
      points, gpack, p2g, N, M);
  // Pass 2: each gt -> nearest point.
  chamfer_rowmin_kernel<<<dim3(M / (32 * WAVES_PER_BLOCK), B), blk, 0, stream>>>(
      gts, ppack, g2p, M, N);

  // Deterministic reductions.
  rowsum_kernel<<<B, 256, 0, stream>>>(p2g, sums, N);
  rowsum_kernel<<<B, 256, 0, stream>>>(g2p, sums + B, M);
  finalize_kernel<<<1, 1, 0, stream>>>(sums, (float*)d_out, B, N, M);
}